// SAGEGraph_20126216749290
// MI455X (gfx1250) — compile-verified
//
#include <hip/hip_runtime.h>

// -------------------------------------------------------------------------
// GraphSAGE 2-layer forward for MI455X (gfx1250, wave32).
//  phase 1: edge scatter (gather x[src], atomic fadd into agg[dst], count)
//  phase 2: fused mean + dual GEMM via V_WMMA_F32_16X16X4_F32
//  phase 3: repeat for layer 2, fusing global mean-pool into the GEMM
//  phase 4: tiny 128x40 head
// Weights live in LDS in a K-pair interleaved layout so every B fragment
// is a single contiguous ds_load_b64 into an even-aligned VGPR pair.
// -------------------------------------------------------------------------

typedef float v2f __attribute__((ext_vector_type(2)));
typedef float v8f __attribute__((ext_vector_type(8)));

__device__ __forceinline__ void atomAddF(float* p, float v) {
  unsafeAtomicAdd(p, v);   // hardware global_atomic_add_f32 (no CAS loop)
}

// ---- scatter: one wave per edge, lane handles 4 consecutive features ----
__global__ __launch_bounds__(256) void scatter_sum_kernel(
    const float* __restrict__ feat, const int* __restrict__ eidx,
    float* __restrict__ agg, float* __restrict__ cnt, int E)
{
  const int wave = (int)((blockIdx.x * blockDim.x + threadIdx.x) >> 5);
  const int lane = threadIdx.x & 31;
  if (wave >= E) return;
  const int s = eidx[wave];       // src
  const int d = eidx[E + wave];   // dst
  const float4 v = *(const float4*)(feat + (size_t)s * 128 + lane * 4);
  float* a = agg + (size_t)d * 128 + lane * 4;
  atomAddF(a + 0, v.x);
  atomAddF(a + 1, v.y);
  atomAddF(a + 2, v.z);
  atomAddF(a + 3, v.w);
  if (cnt != nullptr && lane == 0) atomAddF(cnt + d, 1.0f);
}

// ---- fused SAGE layer: H = mean(AGG) @ Wl + X @ Wr + b ------------------
// One wave owns a 16-node tile and all 128 output columns (8 accum tiles).
// One 128x128 weight matrix staged in LDS at a time (64 KB), interleaved:
//   ldsW[(p*128 + c)*2 + r] = W[2p + r][c]   (p = K-pair row, r in {0,1})
__global__ __launch_bounds__(256) void sage_wmma_kernel(
    const float* __restrict__ X, const float* __restrict__ AGG,
    const float* __restrict__ CNT,
    const float* __restrict__ Wl, const float* __restrict__ Wr,
    const float* __restrict__ bias,
    float* __restrict__ H, float* __restrict__ gsum, int nTiles)
{
  __shared__ float ldsW[128 * 128];   // 64 KB: one weight matrix (interleaved)

  const int tid  = threadIdx.x;
  const int wave = tid >> 5;
  const int lane = tid & 31;
  const int l15  = lane & 15;
  const int half = lane >> 4;

  const int  tile    = blockIdx.x * 8 + wave;
  const bool active  = tile < nTiles;
  const int  rowBase = tile * 16;
  const int  myRow   = rowBase + l15;       // A-fragment row for this lane

  float inv = 1.0f;
  if (active) inv = 1.0f / fmaxf(CNT[myRow], 1.0f);

  v8f acc[8];
  #pragma unroll
  for (int nt = 0; nt < 8; ++nt) {
    const float b = active ? bias[nt * 16 + l15] : 0.0f;
    #pragma unroll
    for (int j = 0; j < 8; ++j) acc[nt][j] = b;   // bias broadcast over rows
  }

  // phase 0: mean(AGG) @ Wl     phase 1: X @ Wr   (same accumulators)
  #pragma unroll
  for (int phase = 0; phase < 2; ++phase) {
    const float* Wg = (phase == 0) ? Wl : Wr;
    __syncthreads();                       // previous phase's LDS reads done
    // cooperative interleaving load: pair rows (2p, 2p+1) of W
    #pragma unroll
    for (int i = 0; i < 8; ++i) {
      const int p = (tid >> 5) + i * 8;    // 0..63
      const int c = (tid & 31) * 4;        // 0..124
      const float4 r0 = *(const float4*)(Wg + (size_t)(2 * p)     * 128 + c);
      const float4 r1 = *(const float4*)(Wg + (size_t)(2 * p + 1) * 128 + c);
      float* o = ldsW + (p * 128 + c) * 2; // 32B-aligned contiguous slot
      const float4 lo = {r0.x, r1.x, r0.y, r1.y};
      const float4 hi = {r0.z, r1.z, r0.w, r1.w};
      *(float4*)(o)     = lo;
      *(float4*)(o + 4) = hi;
    }
    __syncthreads();

    if (active) {
      const float* Ap = ((phase == 0) ? AGG : X) + (size_t)myRow * 128 + half * 2;
      const float scale = (phase == 0) ? inv : 1.0f;
      for (int k = 0; k < 32; ++k) {       // K = 128 in steps of 4
        v2f a = *(const v2f*)(Ap + k * 4); // A[myRow][4k+2*half .. +1]
        a.x *= scale; a.y *= scale;
        // kk = 4k + 2*half  ->  pair row p = 2k + half
        const float* Bp = ldsW + ((2 * k + half) * 128 + l15) * 2;
        #pragma unroll
        for (int nt = 0; nt < 8; ++nt) {   // reuse A fragment across 8 tiles
          const v2f b = *(const v2f*)(Bp + nt * 32); // (W[kk][c], W[kk+1][c])
          acc[nt] = __builtin_amdgcn_wmma_f32_16x16x4_f32(
              false, a, false, b, (short)0, acc[nt], false, false);
        }
      }
    }
  }

  if (!active) return;

  if (H != nullptr) {
    #pragma unroll
    for (int nt = 0; nt < 8; ++nt) {
      float* Hp = H + (size_t)(rowBase + half * 8) * 128 + nt * 16 + l15;
      #pragma unroll
      for (int j = 0; j < 8; ++j)          // C/D layout: vgpr j = row j+8*half
        Hp[(size_t)j * 128] = acc[nt][j];
    }
  }
  if (gsum != nullptr) {                   // fused global mean-pool (sum part)
    #pragma unroll
    for (int nt = 0; nt < 8; ++nt) {
      float s = 0.0f;
      #pragma unroll
      for (int j = 0; j < 8; ++j) s += acc[nt][j];
      atomAddF(gsum + nt * 16 + l15, s);   // per-lane = 8 rows of one column
    }
  }
}

// ---- head: out[c] = b_out[c] + sum_f (gsum[f]/N) * W_out[f][c] ----------
__global__ __launch_bounds__(64) void head_kernel(
    const float* __restrict__ gsum, const float* __restrict__ Wout,
    const float* __restrict__ bout, float* __restrict__ out,
    int classes, float invN)
{
  const int c = threadIdx.x;
  if (c >= classes) return;
  float a = bout[c];
  for (int f = 0; f < 128; ++f)
    a += gsum[f] * invN * Wout[f * classes + c];
  out[c] = a;
}

extern "C" void kernel_launch(void* const* d_in, const int* in_sizes, int n_in,
                              void* d_out, int out_size, void* d_ws, size_t ws_size,
                              hipStream_t stream)
{
  const float* x    = (const float*)d_in[0];
  const int*   eidx = (const int*)d_in[1];
  const float* W1l  = (const float*)d_in[2];
  const float* W1r  = (const float*)d_in[3];
  const float* b1   = (const float*)d_in[4];
  const float* W2l  = (const float*)d_in[5];
  const float* W2r  = (const float*)d_in[6];
  const float* b2   = (const float*)d_in[7];
  const float* Wout = (const float*)d_in[8];
  const float* bout = (const float*)d_in[9];

  const int n       = in_sizes[0] / 128;   // 100000
  const int E       = in_sizes[1] / 2;     // 1600000
  const int classes = in_sizes[9];         // 40

  float* agg  = (float*)d_ws;                    // n*128 (reused both layers)
  float* h1   = agg + (size_t)n * 128;           // n*128
  float* cnt  = h1  + (size_t)n * 128;           // n
  float* gsum = cnt + n;                         // 128

  const int nTiles     = (n + 15) / 16;
  const int gemmBlocks = (nTiles + 7) / 8;
  const int scatBlocks = (E + 7) / 8;            // 8 edges (waves) per block

  hipMemsetAsync(agg,  0, (size_t)n * 128 * sizeof(float), stream);
  hipMemsetAsync(cnt,  0, (size_t)n * sizeof(float), stream);
  hipMemsetAsync(gsum, 0, 128 * sizeof(float), stream);

  // layer 1
  scatter_sum_kernel<<<scatBlocks, 256, 0, stream>>>(x, eidx, agg, cnt, E);
  sage_wmma_kernel<<<gemmBlocks, 256, 0, stream>>>(x, agg, cnt, W1l, W1r, b1,
                                                   h1, nullptr, nTiles);
  // layer 2 (degree counts unchanged; h2 never materialized)
  hipMemsetAsync(agg, 0, (size_t)n * 128 * sizeof(float), stream);
  scatter_sum_kernel<<<scatBlocks, 256, 0, stream>>>(h1, eidx, agg, nullptr, E);
  sage_wmma_kernel<<<gemmBlocks, 256, 0, stream>>>(h1, agg, cnt, W2l, W2r, b2,
                                                   nullptr, gsum, nTiles);
  // pooled head
  head_kernel<<<1, 64, 0, stream>>>(gsum, Wout, bout, (float*)d_out,
                                    classes, 1.0f / (float)n);
}